// AGDLoss_30777735643521
// MI455X (gfx1250) — compile-verified
//
#include <hip/hip_runtime.h>
#include <hip/hip_bf16.h>
#include <math.h>

// ---------------------------------------------------------------------------
// AGD (angular-Gaussian-density) contrastive loss, CDNA5/gfx1250.
//   n = 32768 samples, 100 classes, 40-term fp64 Horner series per (c, j).
//   Constant prefactor K cancels in log-ratio; coefficients generated on
//   device via the two-step recurrence c_n = c_{n-2}*(d+n-2)/(n(n-1)).
//   Per-block 128x100 f32 tile staged via the Tensor Data Mover (TDM),
//   synchronized with s_wait_tensorcnt; fp64 compute from LDS.
// ---------------------------------------------------------------------------

#define N_CLS        100
#define DIMF         128
#define NTERMS       40
#define TILE_ROWS    128
#define BLOCK_THREADS 128
#define NVIEWS       2
#define BATCH        16384

typedef unsigned int v4u __attribute__((ext_vector_type(4)));
typedef int          v4i __attribute__((ext_vector_type(4)));
typedef int          v8i __attribute__((ext_vector_type(8)));

__global__ void agd_tile_kernel(const float* __restrict__ features,
                                const int*   __restrict__ labels,
                                double*      __restrict__ partials) {
  extern __shared__ float tile[];          // TILE_ROWS * N_CLS floats (dynamic LDS)
  __shared__ double sCoef[NTERMS];
  __shared__ double sPart[BLOCK_THREADS];

  const int tid = threadIdx.x;
  const int b0  = blockIdx.x * TILE_ROWS;  // batch-index base of this tile
  const int vv  = blockIdx.y;              // view index

  if (tid == 0) {
    // ---- Issue TDM load: 128 rows x 100 f32, global row stride 256 f32. ----
    // Rows are features[b, vv, 0:100]; consecutive b are 2*128 floats apart.
    unsigned lds_off = __builtin_amdgcn_groupstaticsize();  // dynamic LDS base
    unsigned long long gaddr =
        (unsigned long long)(const void*)(features + ((size_t)(b0 * NVIEWS + vv)) * DIMF);

    v4u g0 = {};
    g0.x = 1u;                                             // count=1 (valid user D#)
    g0.y = lds_off;                                        // lds_addr (bytes)
    g0.z = (unsigned)(gaddr & 0xFFFFFFFFu);                // global_addr[31:0]
    g0.w = (unsigned)((gaddr >> 32) & 0x01FFFFFFu)         // global_addr[56:32]
         | (2u << 30);                                     // type = 2 ("image")

    v8i g1;
    g1[0] = (int)(2u << 16);          // data_size = 2 -> 4 bytes/elem
    g1[1] = (int)(DIMF << 16);        // tensor_dim0 = 128 elems (bits 63:48)
    g1[2] = (int)(TILE_ROWS << 16);   // tensor_dim1 = 128 rows (bits 95:80)
    g1[3] = (int)(N_CLS << 16);       // tile_dim0  = 100 elems (bits 127:112)
    g1[4] = TILE_ROWS;                // tile_dim1  = 128 rows  (bits 143:128)
    g1[5] = NVIEWS * DIMF;            // tensor_dim0_stride = 256 elems (bits 191:160)
    g1[6] = 0;                        // stride hi / dim1_stride lo
    g1[7] = 0;

    v4i z4 = {0, 0, 0, 0};
#if defined(__clang_major__) && (__clang_major__ >= 23)
    v8i z8 = {0, 0, 0, 0, 0, 0, 0, 0};
    __builtin_amdgcn_tensor_load_to_lds(g0, g1, z4, z4, z8, 0);
#else
    __builtin_amdgcn_tensor_load_to_lds(g0, g1, z4, z4, 0);
#endif

    // ---- Saw-series coefficients in fp64 (overlaps the DMA). ----
    // c0 = 1 ; c1 = sqrt(2) * Gamma(64.5)/Gamma(64)
    //        = sqrt(2) * 0.5*sqrt(pi) * prod_{j=1..63} (j+0.5)/j
    double r = 0.5 * sqrt(M_PI);
    for (int j = 1; j <= 63; ++j) r *= ((double)j + 0.5) / (double)j;
    sCoef[0] = 1.0;
    sCoef[1] = sqrt(2.0) * r;
    for (int nn = 2; nn < NTERMS; ++nn)
      sCoef[nn] = sCoef[nn - 2] * (double)(DIMF + nn - 2)
                / ((double)nn * (double)(nn - 1));

    __builtin_amdgcn_s_wait_tensorcnt(0);   // TDM tile resident in LDS
  }
  __syncthreads();

  // ---- fp64 Horner over 100 classes, 4 independent chains for ILP. ----
  const float* yrow = &tile[tid * N_CLS];
  const int lab = labels[b0 + tid];         // labels_rep[v*B + b] == labels[b]

  double norm = 0.0;
  double slab = 1.0;
  const double ctop = sCoef[NTERMS - 1];
  for (int c = 0; c < N_CLS; c += 4) {
    double a0 = (double)yrow[c + 0];
    double a1 = (double)yrow[c + 1];
    double a2 = (double)yrow[c + 2];
    double a3 = (double)yrow[c + 3];
    double s0 = ctop, s1 = ctop, s2 = ctop, s3 = ctop;
    for (int k = NTERMS - 2; k >= 0; --k) {
      double ck = sCoef[k];
      s0 = fma(s0, a0, ck);
      s1 = fma(s1, a1, ck);
      s2 = fma(s2, a2, ck);
      s3 = fma(s3, a3, ck);
    }
    norm += (s0 + s1) + (s2 + s3);
    if (lab == c + 0) slab = s0;
    if (lab == c + 1) slab = s1;
    if (lab == c + 2) slab = s2;
    if (lab == c + 3) slab = s3;
  }

  // loss_j = log(norms_j) - log(dens_lab_j); the constant K cancels.
  sPart[tid] = log(norm) - log(slab);
  __syncthreads();

  // Deterministic fixed-shape tree reduction.
  for (int off = BLOCK_THREADS / 2; off > 0; off >>= 1) {
    if (tid < off) sPart[tid] += sPart[tid + off];
    __syncthreads();
  }
  if (tid == 0)
    partials[blockIdx.y * gridDim.x + blockIdx.x] = sPart[0];
}

__global__ void agd_reduce_kernel(const double* __restrict__ partials,
                                  float* __restrict__ out, int nparts) {
  // Single thread, fixed order -> bitwise deterministic across replays.
  double t = 0.0;
  for (int i = 0; i < nparts; ++i) t += partials[i];
  out[0] = (float)t;
}

extern "C" void kernel_launch(void* const* d_in, const int* in_sizes, int n_in,
                              void* d_out, int out_size, void* d_ws, size_t ws_size,
                              hipStream_t stream) {
  (void)in_sizes; (void)n_in; (void)out_size; (void)ws_size;
  const float* features = (const float*)d_in[0];   // [B, NVIEWS, DIMF] f32
  const int*   labels   = (const int*)d_in[1];     // [B] int
  double* partials = (double*)d_ws;                // 256 fp64 partials

  dim3 grid(BATCH / TILE_ROWS, NVIEWS);            // (128, 2)
  size_t shmem = (size_t)TILE_ROWS * N_CLS * sizeof(float);  // 51200 B
  agd_tile_kernel<<<grid, BLOCK_THREADS, shmem, stream>>>(features, labels, partials);

  int nparts = (BATCH / TILE_ROWS) * NVIEWS;       // 256
  agd_reduce_kernel<<<1, 1, 0, stream>>>(partials, (float*)d_out, nparts);
}